// Net_31576599560691
// MI455X (gfx1250) — compile-verified
//
#include <hip/hip_runtime.h>
#include <hip/hip_bf16.h>

typedef float v2f __attribute__((ext_vector_type(2)));
typedef float v8f __attribute__((ext_vector_type(8)));

constexpr int N_NODES  = 100000;
constexpr int N_EDGES  = 600000;
constexpr int N_GRAPHS = 64;
constexpr int D_IN     = 128;
constexpr int D_H1     = 100;
constexpr int D_H2     = 20;

// ---------------------------------------------------------------- utilities

__global__ void zero_f32(float* __restrict__ p, int n) {
  int i = blockIdx.x * blockDim.x + threadIdx.x;
  int stride = gridDim.x * blockDim.x;
  for (; i < n; i += stride) p[i] = 0.0f;
}

__global__ void deg_kernel(const int* __restrict__ dst, float* __restrict__ deg) {
  int i = blockIdx.x * blockDim.x + threadIdx.x;
  if (i < N_EDGES) atomicAdd(&deg[dst[i]], 1.0f);
}

__global__ void cnt_kernel(const int* __restrict__ gid, float* __restrict__ cnt) {
  int i = blockIdx.x * blockDim.x + threadIdx.x;
  if (i < N_NODES) atomicAdd(&cnt[gid[i]], 1.0f);
}

__global__ void inv_kernel(const float* __restrict__ deg, float* __restrict__ inv) {
  int i = blockIdx.x * blockDim.x + threadIdx.x;
  if (i < N_NODES) inv[i] = 1.0f / fmaxf(deg[i], 1.0f);
}

// Scatter-add projected features along edges: agg[dst] += P[src], D floats each.
// 32 lanes cooperate per edge (coalesced within the row).
template <int D>
__global__ void edge_agg(const float* __restrict__ P,
                         const int* __restrict__ src,
                         const int* __restrict__ dst,
                         float* __restrict__ agg) {
  unsigned tid = blockIdx.x * blockDim.x + threadIdx.x;
  int e = (int)(tid >> 5);
  if (e >= N_EDGES) return;
  int lane = (int)(tid & 31u);
  const float* ps = P + (size_t)src[e] * D;
  float* pd = agg + (size_t)dst[e] * D;
  for (int idx = lane; idx < D; idx += 32) atomicAdd(&pd[idx], ps[idx]);
}

// ---------------------------------------------------------------- WMMA GEMM
// Computes a 16-row strip of A[N,K] @ W[K,NCOLS] with f32 WMMA (16x16x4).
// MODE 0: out[m,n] = acc                              (neighbor projection)
// MODE 1: out[m,n] = relu(acc + inv[m]*agg[m,n]+b[n]) (layer-1 self, fused)
// MODE 2: atomicAdd(hg[gid[m],n], relu(...))          (layer-2 self + pooling)
template <int K, int NCOLS, int MODE>
__global__ void __launch_bounds__(128)
sage_gemm(const float* __restrict__ A,
          const float* __restrict__ W,
          const float* __restrict__ bias,
          const float* __restrict__ agg,
          const float* __restrict__ invdeg,
          const int* __restrict__ gid,
          float* __restrict__ out) {
  constexpr int NT = (NCOLS + 15) / 16;
  const int wave = (int)((blockIdx.x * blockDim.x + threadIdx.x) >> 5);
  const int m0 = wave * 16;
  if (m0 >= N_NODES) return;               // wave-uniform: EXEC stays all-ones
  const int lane = (int)(threadIdx.x & 31u);
  const int half = lane >> 4;              // 0: K=k,k+1   1: K=k+2,k+3
  const int l16 = lane & 15;
  const float* Arow = A + (size_t)(m0 + l16) * K;

  v8f c[NT];
#pragma unroll
  for (int t = 0; t < NT; ++t) c[t] = (v8f)0.0f;

  for (int k = 0; k < K; k += 4) {
    const int kb = k + 2 * half;
    v2f a = *(const v2f*)(Arow + kb);      // 8B-aligned for K=128 and K=100
#pragma unroll
    for (int t = 0; t < NT; ++t) {
      int n = t * 16 + l16;
      if (n > NCOLS - 1) n = NCOLS - 1;    // clamp ragged tile (masked at store)
      v2f b;
      b.x = W[kb * NCOLS + n];
      b.y = W[(kb + 1) * NCOLS + n];
      c[t] = __builtin_amdgcn_wmma_f32_16x16x4_f32(
          false, a, false, b, (short)0, c[t], false, false);
    }
  }

#pragma unroll
  for (int t = 0; t < NT; ++t) {
    const int n = t * 16 + l16;
    if (n >= NCOLS) continue;
#pragma unroll
    for (int r = 0; r < 8; ++r) {
      const int m = m0 + r + 8 * half;
      float v = c[t][r];
      if (MODE == 0) {
        out[(size_t)m * NCOLS + n] = v;
      } else {
        v += invdeg[m] * agg[(size_t)m * NCOLS + n] + bias[n];
        v = fmaxf(v, 0.0f);
        if (MODE == 1) {
          out[(size_t)m * NCOLS + n] = v;
        } else {
          atomicAdd(&out[gid[m] * NCOLS + n], v);
        }
      }
    }
  }
}

// ---------------------------------------------------------------- final MLP
__global__ void final_mlp(const float* __restrict__ hg, const float* __restrict__ cnt,
                          const float* __restrict__ fc1_w, const float* __restrict__ fc1_b,
                          const float* __restrict__ fc2_w, const float* __restrict__ fc2_b,
                          float* __restrict__ out) {
  int g = threadIdx.x;
  if (g >= N_GRAPHS) return;
  float inv = 1.0f / fmaxf(cnt[g], 1.0f);
  float mean[D_H2];
#pragma unroll
  for (int n = 0; n < D_H2; ++n) mean[n] = hg[g * D_H2 + n] * inv;
  float o = fc2_b[0];
#pragma unroll
  for (int j = 0; j < 10; ++j) {
    float h = fc1_b[j];
#pragma unroll
    for (int n = 0; n < D_H2; ++n) h += mean[n] * fc1_w[n * 10 + j];
    o += fmaxf(h, 0.0f) * fc2_w[j];
  }
  out[g] = o;
}

// ---------------------------------------------------------------- launcher
extern "C" void kernel_launch(void* const* d_in, const int* in_sizes, int n_in,
                              void* d_out, int out_size, void* d_ws, size_t ws_size,
                              hipStream_t stream) {
  (void)in_sizes; (void)n_in; (void)out_size; (void)ws_size;
  const float* feat    = (const float*)d_in[0];
  const float* Wself1  = (const float*)d_in[1];
  const float* Wneigh1 = (const float*)d_in[2];
  const float* b1      = (const float*)d_in[3];
  const float* Wself2  = (const float*)d_in[4];
  const float* Wneigh2 = (const float*)d_in[5];
  const float* b2      = (const float*)d_in[6];
  const float* fc1_w   = (const float*)d_in[7];
  const float* fc1_b   = (const float*)d_in[8];
  const float* fc2_w   = (const float*)d_in[9];
  const float* fc2_b   = (const float*)d_in[10];
  const int*   src     = (const int*)d_in[11];
  const int*   dst     = (const int*)d_in[12];
  const int*   gid     = (const int*)d_in[13];
  float* out = (float*)d_out;

  // workspace layout (floats): deg | inv | bufA(N*100) | bufB(N*100) | hg | cnt
  float* ws   = (float*)d_ws;
  float* deg  = ws;
  float* inv  = deg + N_NODES;
  float* bufA = inv + N_NODES;                       // Pneigh1, later h1
  float* bufB = bufA + (size_t)N_NODES * D_H1;       // agg1, later Pneigh2 | agg2
  float* hg   = bufB + (size_t)N_NODES * D_H1;       // 64*20 graph sums
  float* cnt  = hg + N_GRAPHS * D_H2;                // 64 node counts

  const int GEMM_BLOCKS = (N_NODES / 16 * 32 + 127) / 128;  // 1563 blocks of 4 waves
  const int EDGE_BLOCKS = (N_EDGES * 32 + 255) / 256;

  // zero accumulators: deg, bufB(agg1), hg+cnt (hg/cnt are contiguous after bufB)
  zero_f32<<<1024, 256, 0, stream>>>(deg, N_NODES);
  zero_f32<<<4096, 256, 0, stream>>>(bufB, N_NODES * D_H1 + N_GRAPHS * D_H2 + N_GRAPHS);

  deg_kernel<<<(N_EDGES + 255) / 256, 256, 0, stream>>>(dst, deg);
  cnt_kernel<<<(N_NODES + 255) / 256, 256, 0, stream>>>(gid, cnt);
  inv_kernel<<<(N_NODES + 255) / 256, 256, 0, stream>>>(deg, inv);

  // ---- layer 1: project-then-aggregate (aggregate in 100-dim, not 128) ----
  sage_gemm<D_IN, D_H1, 0><<<GEMM_BLOCKS, 128, 0, stream>>>(
      feat, Wneigh1, nullptr, nullptr, nullptr, nullptr, bufA);       // Pneigh1
  edge_agg<D_H1><<<EDGE_BLOCKS, 256, 0, stream>>>(bufA, src, dst, bufB);  // agg1
  sage_gemm<D_IN, D_H1, 1><<<GEMM_BLOCKS, 128, 0, stream>>>(
      feat, Wself1, b1, bufB, inv, nullptr, bufA);                    // h1 (overwrites Pneigh1)

  // ---- layer 2: aggregate in 20-dim ----
  float* Pneigh2 = bufB;
  float* agg2    = bufB + (size_t)N_NODES * D_H2;
  zero_f32<<<2048, 256, 0, stream>>>(agg2, N_NODES * D_H2);
  sage_gemm<D_H1, D_H2, 0><<<GEMM_BLOCKS, 128, 0, stream>>>(
      bufA, Wneigh2, nullptr, nullptr, nullptr, nullptr, Pneigh2);
  edge_agg<D_H2><<<EDGE_BLOCKS, 256, 0, stream>>>(Pneigh2, src, dst, agg2);
  sage_gemm<D_H1, D_H2, 2><<<GEMM_BLOCKS, 128, 0, stream>>>(
      bufA, Wself2, b2, agg2, inv, gid, hg);                          // h2 fused into graph sums

  // ---- per-graph mean + tiny MLP ----
  final_mlp<<<1, 64, 0, stream>>>(hg, cnt, fc1_w, fc1_b, fc2_w, fc2_b, out);
}